// DualGATConv_60464549593447
// MI455X (gfx1250) — compile-verified
//
#include <hip/hip_runtime.h>
#include <hip/hip_bf16.h>

typedef __attribute__((ext_vector_type(2))) float v2f;
typedef __attribute__((ext_vector_type(8))) float v8f;

#define NEG_SLOPE 0.2f

// ---------- order-preserving float <-> uint encoding for atomicMax ----------
__device__ __forceinline__ unsigned fenc(float f) {
    unsigned i = __float_as_uint(f);
    return (i & 0x80000000u) ? ~i : (i | 0x80000000u);
}
__device__ __forceinline__ float fdec(unsigned u) {
    return (u & 0x80000000u) ? __uint_as_float(u & 0x7fffffffu)
                             : __uint_as_float(~u);
}

// ---------------------------------------------------------------------------
// C[M,NC] = A[M,128] @ B[128,NC] (+ bias[col] if bias != nullptr), row-major.
// NC is compile-time so all B/A offsets fold into immediate load offsets.
// One wave32 per 16x(16*NT) output strip: per k-step, load the A fragment and
// all NT B fragments into distinct registers (one load clause), then issue NT
// independent v_wmma_f32_16x16x4_f32 back-to-back (latency hiding).
// A frag (16x4 f32): VGPR0 = K(half*2+0), VGPR1 = K(half*2+1); lanes 0-15 M,
// lanes 16-31 use K offset +2.  B frag (4x16) mirrored.
// ---------------------------------------------------------------------------
template <int NT, int NC>
__global__ __launch_bounds__(32) void gemm128_wmma(
    const float* __restrict__ A, const float* __restrict__ B,
    const float* __restrict__ bias, float* __restrict__ C, int M) {
    constexpr int strips = NC / (16 * NT);
    const int tm = blockIdx.x / strips;
    const int ts = blockIdx.x % strips;
    const int lane = threadIdx.x;
    const int half = lane >> 4;   // 0 or 1
    const int l16  = lane & 15;

    int rowA = tm * 16 + l16;
    int rA = rowA < M ? rowA : (M - 1);     // clamp loads, predicate stores
    const int k0base = half * 2;
    const float* Ap = A + (size_t)rA * 128 + k0base;      // per-wave A base
    const int col0 = ts * 16 * NT + l16;
    const float* Bp = B + (size_t)k0base * NC + col0;     // per-wave B base

    const v8f vzero = {0.f, 0.f, 0.f, 0.f, 0.f, 0.f, 0.f, 0.f};
    v8f c[NT];
#pragma unroll
    for (int t = 0; t < NT; ++t) c[t] = vzero;

#pragma unroll
    for (int kk = 0; kk < 128; kk += 4) {
        v2f a;
        a.x = Ap[kk];                 // A[rA, kk + half*2 + 0]
        a.y = Ap[kk + 1];             // A[rA, kk + half*2 + 1]
        v2f b[NT];
#pragma unroll
        for (int t = 0; t < NT; ++t) {      // one clause of NT*2 loads
            b[t].x = Bp[(size_t)kk * NC + t * 16];
            b[t].y = Bp[(size_t)(kk + 1) * NC + t * 16];
        }
#pragma unroll
        for (int t = 0; t < NT; ++t) {      // NT independent WMMA chains
            c[t] = __builtin_amdgcn_wmma_f32_16x16x4_f32(
                false, a, false, b[t], (short)0, c[t], false, false);
        }
    }
#pragma unroll
    for (int t = 0; t < NT; ++t) {
        float bv = bias ? bias[col0 + t * 16] : 0.f;
#pragma unroll
        for (int r = 0; r < 8; ++r) {
            int row = tm * 16 + r + half * 8;
            if (row < M) C[(size_t)row * NC + col0 + t * 16] = c[t][r] + bv;
        }
    }
}

// ---------------- zero / init accumulators ----------------
__global__ void init_gat_kernel(float* acc, float* s, unsigned* m, long long n) {
    long long gid = (long long)blockIdx.x * blockDim.x + threadIdx.x;
    if (gid >= n * 128) return;
    acc[gid] = 0.f;
    if (gid < n * 4) { s[gid] = 0.f; m[gid] = 0u; }
}

__global__ void init_adj_kernel(float* adj, float* cnt, long long g) {
    long long gid = (long long)blockIdx.x * blockDim.x + threadIdx.x;
    if (gid >= g * 128) return;
    adj[gid] = 0.f;
    if (gid < g) cnt[gid] = 0.f;
}

// ---------------- attention coefficients: a_src/a_dst [n,4] ----------------
__global__ void attn_kernel(const float* __restrict__ h,
                            const float* __restrict__ ws,
                            const float* __restrict__ wd,
                            float* __restrict__ a_s, float* __restrict__ a_d,
                            int n) {
    long long gid = (long long)blockIdx.x * blockDim.x + threadIdx.x;
    if (gid >= (long long)n * 4) return;
    int node = (int)(gid >> 2), hd = (int)(gid & 3);
    const float* hr = h + (size_t)node * 128 + hd * 32;
    float ss = 0.f, sd = 0.f;
#pragma unroll
    for (int c = 0; c < 32; ++c) {
        float hv = hr[c];
        ss += hv * ws[hd * 32 + c];
        sd += hv * wd[hd * 32 + c];
    }
    a_s[gid] = ss;
    a_d[gid] = sd;
}

// ---------------- edge pass 1: segment max of leaky_relu(e) ----------------
__global__ void edge_max_kernel(const int* __restrict__ ei, int E,
                                const float* __restrict__ a_s,
                                const float* __restrict__ a_d,
                                unsigned* __restrict__ m, long long Etot) {
    long long gid = (long long)blockIdx.x * blockDim.x + threadIdx.x;
    if (gid >= Etot * 4) return;
    int e = (int)(gid >> 2), hd = (int)(gid & 3);
    int src, dst;
    if (e < E) { src = ei[e]; dst = ei[E + e]; } else { src = dst = e - E; }
    float v = a_s[src * 4 + hd] + a_d[dst * 4 + hd];
    v = v > 0.f ? v : NEG_SLOPE * v;
    atomicMax(&m[dst * 4 + hd], fenc(v));
}

// ------- edge pass 2: s[dst,h] += ex ; acc[dst,ch] += ex * h[src,ch] -------
__global__ void edge_accum_kernel(const int* __restrict__ ei, int E,
                                  const float* __restrict__ a_s,
                                  const float* __restrict__ a_d,
                                  const unsigned* __restrict__ m,
                                  const float* __restrict__ h,
                                  float* __restrict__ s,
                                  float* __restrict__ acc, long long Etot) {
    long long gid = (long long)blockIdx.x * blockDim.x + threadIdx.x;
    if (gid >= Etot * 128) return;
    int e = (int)(gid >> 7), ch = (int)(gid & 127), hd = ch >> 5;
    int src, dst;
    if (e < E) { src = ei[e]; dst = ei[E + e]; } else { src = dst = e - E; }
    float v = a_s[src * 4 + hd] + a_d[dst * 4 + hd];
    v = v > 0.f ? v : NEG_SLOPE * v;
    float ex = __expf(v - fdec(m[dst * 4 + hd]));
    if ((ch & 31) == 0) atomicAdd(&s[dst * 4 + hd], ex);
    atomicAdd(&acc[(size_t)dst * 128 + ch], ex * h[(size_t)src * 128 + ch]);
}

// ---------------- finalize: dst = acc / (s + eps) + bias ----------------
__global__ void finalize_kernel(const float* __restrict__ acc,
                                const float* __restrict__ s,
                                const float* __restrict__ bias,
                                float* __restrict__ dst, long long n) {
    long long gid = (long long)blockIdx.x * blockDim.x + threadIdx.x;
    if (gid >= n * 128) return;
    int node = (int)(gid >> 7), ch = (int)(gid & 127), hd = ch >> 5;
    dst[gid] = acc[gid] / (s[node * 4 + hd] + 1e-16f) + bias[ch];
}

// ------------- adjacency accumulation over group edges -------------
__global__ void adj_accum_kernel(const int* __restrict__ eig, int EG,
                                 const float* __restrict__ xg,
                                 float* __restrict__ adj,
                                 float* __restrict__ cnt) {
    long long gid = (long long)blockIdx.x * blockDim.x + threadIdx.x;
    if (gid >= (long long)EG * 128) return;
    int e = (int)(gid >> 7), ch = (int)(gid & 127);
    int g0 = eig[e], g1 = eig[EG + e];
    atomicAdd(&adj[(size_t)g0 * 128 + ch], xg[(size_t)g1 * 128 + ch]);
    if (ch == 0) atomicAdd(&cnt[g0], 1.f);
}

__global__ void adj_mean_kernel(float* __restrict__ adj,
                                const float* __restrict__ cnt, long long g) {
    long long gid = (long long)blockIdx.x * blockDim.x + threadIdx.x;
    if (gid >= g * 128) return;
    adj[gid] = adj[gid] / fmaxf(cnt[gid >> 7], 1.f);
}

// -------- per-node importance fusion: one wave32 per node (shuffles) --------
__global__ __launch_bounds__(128) void node_update_kernel(
    const float* __restrict__ xo, const int* __restrict__ ng,
    const float* __restrict__ xg, const float* __restrict__ am,
    float* __restrict__ upd, int n) {
    int wave = threadIdx.x >> 5;
    int lane = threadIdx.x & 31;
    int node = blockIdx.x * (blockDim.x >> 5) + wave;
    if (node >= n) return;
    int g = ng[node];
    const float* xr = xo + (size_t)node * 128;
    const float* gf = xg + (size_t)g * 128;
    const float* af = am + (size_t)g * 128;
    float xv[4], gv[4], av[4];
    float pg = 0.f, pa = 0.f;
#pragma unroll
    for (int i = 0; i < 4; ++i) {
        int c = lane + i * 32;
        xv[i] = xr[c]; gv[i] = gf[c]; av[i] = af[c];
        pg += xv[i] * gv[i];
        pa += xv[i] * av[i];
    }
#pragma unroll
    for (int off = 16; off >= 1; off >>= 1) {   // wave32 butterfly reduction
        pg += __shfl_xor(pg, off, 32);
        pa += __shfl_xor(pa, off, 32);
    }
    float* ur = upd + (size_t)node * 128;
#pragma unroll
    for (int i = 0; i < 4; ++i) {
        int c = lane + i * 32;
        ur[c] = xv[i] + pg * gv[i] + pa * av[i];
    }
}

// ---------------------------------------------------------------------------
static inline unsigned cdivll(long long a, long long b) {
    return (unsigned)((a + b - 1) / b);
}

extern "C" void kernel_launch(void* const* d_in, const int* in_sizes, int n_in,
                              void* d_out, int out_size, void* d_ws,
                              size_t ws_size, hipStream_t stream) {
    const float* x      = (const float*)d_in[0];
    const float* x_g    = (const float*)d_in[1];
    const int*   ei     = (const int*)d_in[2];
    const int*   eig    = (const int*)d_in[3];
    const int*   ngrp   = (const int*)d_in[4];
    const float* W1     = (const float*)d_in[5];
    const float* atts1  = (const float*)d_in[6];
    const float* attd1  = (const float*)d_in[7];
    const float* bias1  = (const float*)d_in[8];
    const float* W2     = (const float*)d_in[9];
    const float* atts2  = (const float*)d_in[10];
    const float* attd2  = (const float*)d_in[11];
    const float* bias2  = (const float*)d_in[12];
    const float* W_fc   = (const float*)d_in[13];
    const float* b_fc   = (const float*)d_in[14];

    const int N  = in_sizes[0] / 128;
    const int G  = in_sizes[1] / 128;
    const int E  = in_sizes[2] / 2;
    const int EG = in_sizes[3] / 2;
    const long long EtN = (long long)E + N;   // edges + self loops
    const long long EtG = (long long)EG + G;

    float* out    = (float*)d_out;
    float* xg_out = out + (size_t)N * 32;     // [G,128] tail of d_out

    // ---- workspace carve-up (floats) ----
    float* ws = (float*)d_ws;
    size_t o = 0;
    float*    h     = ws + o; o += (size_t)N * 128;   // h, later upd
    float*    acc   = ws + o; o += (size_t)N * 128;   // acc, later x_out
    float*    a_s   = ws + o; o += (size_t)N * 4;
    float*    a_d   = ws + o; o += (size_t)N * 4;
    float*    sden  = ws + o; o += (size_t)N * 4;
    unsigned* menc  = (unsigned*)(ws + o); o += (size_t)N * 4;
    float*    h_g   = ws + o; o += (size_t)G * 128;
    float*    acc_g = ws + o; o += (size_t)G * 128;
    float*    a_sg  = ws + o; o += (size_t)G * 4;
    float*    a_dg  = ws + o; o += (size_t)G * 4;
    float*    s_g   = ws + o; o += (size_t)G * 4;
    unsigned* m_g   = (unsigned*)(ws + o); o += (size_t)G * 4;
    float*    adj   = ws + o; o += (size_t)G * 128;
    float*    cnt   = ws + o; o += (size_t)G;

    const int TB = 256;

    // ================= big-graph GAT =================
    init_gat_kernel<<<cdivll((long long)N * 128, TB), TB, 0, stream>>>(
        acc, sden, menc, N);
    {   // h = x @ W1   (M=N, NC=128): 16x64 strips -> 2 strips per row tile
        unsigned tiles = (unsigned)(((N + 15) / 16) * (128 / 64));
        gemm128_wmma<4, 128><<<tiles, 32, 0, stream>>>(x, W1, nullptr, h, N);
    }
    attn_kernel<<<cdivll((long long)N * 4, TB), TB, 0, stream>>>(
        h, atts1, attd1, a_s, a_d, N);
    edge_max_kernel<<<cdivll(EtN * 4, TB), TB, 0, stream>>>(
        ei, E, a_s, a_d, menc, EtN);
    edge_accum_kernel<<<cdivll(EtN * 128, TB), TB, 0, stream>>>(
        ei, E, a_s, a_d, menc, h, sden, acc, EtN);
    finalize_kernel<<<cdivll((long long)N * 128, TB), TB, 0, stream>>>(
        acc, sden, bias1, acc, N);                    // acc becomes x_out

    // ================= group-graph GAT =================
    init_gat_kernel<<<cdivll((long long)G * 128, TB), TB, 0, stream>>>(
        acc_g, s_g, m_g, G);
    {   // h_g = x_g @ W2  (M=G, NC=128)
        unsigned tiles = (unsigned)(((G + 15) / 16) * (128 / 64));
        gemm128_wmma<4, 128><<<tiles, 32, 0, stream>>>(x_g, W2, nullptr, h_g, G);
    }
    attn_kernel<<<cdivll((long long)G * 4, TB), TB, 0, stream>>>(
        h_g, atts2, attd2, a_sg, a_dg, G);
    edge_max_kernel<<<cdivll(EtG * 4, TB), TB, 0, stream>>>(
        eig, EG, a_sg, a_dg, m_g, EtG);
    edge_accum_kernel<<<cdivll(EtG * 128, TB), TB, 0, stream>>>(
        eig, EG, a_sg, a_dg, m_g, h_g, s_g, acc_g, EtG);
    finalize_kernel<<<cdivll((long long)G * 128, TB), TB, 0, stream>>>(
        acc_g, s_g, bias2, xg_out, G);                // x_g_out -> d_out tail

    // ================= adjacency-group mean =================
    init_adj_kernel<<<cdivll((long long)G * 128, TB), TB, 0, stream>>>(
        adj, cnt, G);
    adj_accum_kernel<<<cdivll((long long)EG * 128, TB), TB, 0, stream>>>(
        eig, EG, xg_out, adj, cnt);
    adj_mean_kernel<<<cdivll((long long)G * 128, TB), TB, 0, stream>>>(
        adj, cnt, G);

    // ================= importance fusion + final FC =================
    node_update_kernel<<<cdivll(N, 4), 128, 0, stream>>>(
        acc, ngrp, xg_out, adj, h, N);                // upd reuses h buffer
    {   // out = upd @ W_fc + b_fc  (M=N, NC=32): one 16x32 strip per row tile
        unsigned tiles = (unsigned)((N + 15) / 16);
        gemm128_wmma<2, 32><<<tiles, 32, 0, stream>>>(h, W_fc, b_fc, out, N);
    }
}